// GRAND_17497696764532
// MI455X (gfx1250) — compile-verified
//
#include <hip/hip_runtime.h>

// Problem constants (from reference setup_inputs)
#define NN 50000
#define NE 1600000
#define DIN 128
#define DHID 128
#define DOUT 64

typedef __attribute__((ext_vector_type(2))) float v2f;
typedef __attribute__((ext_vector_type(8))) float v8f;

// ---------------------------------------------------------------------------
// Degree / normalization (graph static -> computed once, reused by 3 layers)
// ---------------------------------------------------------------------------
__global__ void deg_init_kernel(float* __restrict__ deg, int n) {
    int i = blockIdx.x * blockDim.x + threadIdx.x;
    if (i < n) deg[i] = 1.0f;  // self-loop contribution
}

__global__ void deg_accum_kernel(const int* __restrict__ src, const float* __restrict__ w,
                                 float* __restrict__ deg, int e) {
    int i = blockIdx.x * blockDim.x + threadIdx.x;
    if (i < e) atomicAdd(&deg[src[i]], w[i]);
}

__global__ void dinv_kernel(float* __restrict__ deg, int n) {
    int i = blockIdx.x * blockDim.x + threadIdx.x;
    if (i < n) {
        float d = deg[i];
        deg[i] = (d > 0.0f) ? rsqrtf(d) : 0.0f;  // in-place: deg -> deg^-1/2
    }
}

// ---------------------------------------------------------------------------
// CSR build (once): histogram -> 2-level exclusive scan -> scatter.
// Turns E*d scatter-atomics per layer into a pure gather (zero atomics).
// ---------------------------------------------------------------------------
__global__ void zero_int_kernel(int* __restrict__ p, int n) {
    int i = blockIdx.x * blockDim.x + threadIdx.x;
    if (i < n) p[i] = 0;
}

__global__ void hist_kernel(const int* __restrict__ src, int* __restrict__ cnt, int e) {
    int i = blockIdx.x * blockDim.x + threadIdx.x;
    if (i < e) atomicAdd(&cnt[src[i]], 1);
}

// Hillis-Steele inclusive scan over 256-element blocks; emit exclusive + block sum
__global__ void scan_blocks_kernel(const int* __restrict__ cnt, int* __restrict__ rowptr,
                                   int* __restrict__ bsums, int n) {
    __shared__ int sm[256];
    int tid = threadIdx.x;
    int i = blockIdx.x * 256 + tid;
    int v = (i < n) ? cnt[i] : 0;
    int val = v;
    sm[tid] = val;
    __syncthreads();
    for (int off = 1; off < 256; off <<= 1) {
        int t = (tid >= off) ? sm[tid - off] : 0;
        __syncthreads();
        val += t;
        sm[tid] = val;
        __syncthreads();
    }
    if (i < n) rowptr[i] = val - v;          // exclusive within block
    if (tid == 255) bsums[blockIdx.x] = val; // block total
}

// Single-block exclusive scan of the (<=256) block sums, in place
__global__ void scan_sums_kernel(int* __restrict__ bsums, int nb) {
    __shared__ int sm[256];
    int tid = threadIdx.x;
    int v = (tid < nb) ? bsums[tid] : 0;
    int val = v;
    sm[tid] = val;
    __syncthreads();
    for (int off = 1; off < 256; off <<= 1) {
        int t = (tid >= off) ? sm[tid - off] : 0;
        __syncthreads();
        val += t;
        sm[tid] = val;
        __syncthreads();
    }
    if (tid < nb) bsums[tid] = val - v;      // exclusive block offsets
}

// Finalize rowptr, replicate into fill[] as the scatter cursor
__global__ void add_off_kernel(int* __restrict__ rowptr, int* __restrict__ fill,
                               const int* __restrict__ bsums, int n, int e) {
    int i = blockIdx.x * blockDim.x + threadIdx.x;
    if (i < n) {
        int r = rowptr[i] + bsums[i >> 8];
        rowptr[i] = r;
        fill[i] = r;
    }
    if (i == 0) rowptr[n] = e;
}

// Scatter edges into CSR order; precompute per-edge coefficient once
__global__ void scatter_kernel(const int* __restrict__ src, const int* __restrict__ dst,
                               const float* __restrict__ w, const float* __restrict__ dinv,
                               int* __restrict__ fill, int* __restrict__ eDst,
                               float* __restrict__ eCoef, int e) {
    int i = blockIdx.x * blockDim.x + threadIdx.x;
    if (i < e) {
        int s = src[i];
        int t = dst[i];
        int pos = atomicAdd(&fill[s], 1);
        eDst[pos] = t;
        eCoef[pos] = dinv[s] * w[i] * dinv[t];
    }
}

// ---------------------------------------------------------------------------
// Dense GEMM via fp32 WMMA: C[M,N] = relu?(H)[M,K] @ W[N,K]^T + bias[N]
// One wave32 per 16x16 output tile; K stepped by 4 (V_WMMA_F32_16X16X4_F32).
// A 16x4 f32 (ISA 7.12.2): lane = 16*lh + m; VGPR0 = K=2*lh, VGPR1 = K=2*lh+1.
// B 4x16 mirrors C striping. C/D: VGPR r -> (M = r + 8*lh, N = lane&15).
// ---------------------------------------------------------------------------
template <bool RELU_IN>
__global__ void gemm_wmma_kernel(const float* __restrict__ H, const float* __restrict__ W,
                                 const float* __restrict__ bias, float* __restrict__ C,
                                 int din, int dout, int total_tiles, int tiles_n) {
    int lane = threadIdx.x & 31;
    int wave = threadIdx.x >> 5;
    int tile = blockIdx.x * (blockDim.x >> 5) + wave;
    if (tile >= total_tiles) return;  // wave-uniform: EXEC stays all-1s for WMMA

    int tm = tile / tiles_n;
    int tn = tile - tm * tiles_n;
    int l16 = lane & 15;
    int lh  = lane >> 4;

    const float* arow = H + (size_t)(tm * 16 + l16) * din;
    const float* brow = W + (size_t)(tn * 16 + l16) * din;

    v8f acc = {};
    for (int k0 = 0; k0 < din; k0 += 4) {
        int kb = k0 + 2 * lh;
        float a0 = arow[kb], a1 = arow[kb + 1];
        if (RELU_IN) { a0 = fmaxf(a0, 0.0f); a1 = fmaxf(a1, 0.0f); }
        v2f a, b;
        a.x = a0;       a.y = a1;
        b.x = brow[kb]; b.y = brow[kb + 1];
        acc = __builtin_amdgcn_wmma_f32_16x16x4_f32(
            false, a, false, b, (short)0, acc, false, false);
    }

    float bv = bias[tn * 16 + l16];
    float* crow = C + (size_t)(tm * 16 + 8 * lh) * dout + (tn * 16 + l16);
#pragma unroll
    for (int r = 0; r < 8; ++r)
        crow[(size_t)r * dout] = acc[r] + bv;
}

// ---------------------------------------------------------------------------
// CSR gather aggregation: out[i,:] = dinv[i]^2*h[i,:] + sum_j coef_j*h[dst_j,:]
// Wave-per-node. Edge (dst,coef) pairs loaded coalesced per lane, broadcast
// with __shfl (ds_bpermute/permlane). Each lane owns a float4 channel chunk,
// accumulates in registers, single store. Zero atomics; gathers hit L2.
// ---------------------------------------------------------------------------
__global__ void agg_gather_kernel(const int* __restrict__ rowptr, const int* __restrict__ eDst,
                                  const float* __restrict__ eCoef, const float* __restrict__ dinv,
                                  const float* __restrict__ h, float* __restrict__ out,
                                  int n, int d) {
    int gid  = blockIdx.x * blockDim.x + threadIdx.x;
    int node = gid >> 5;
    int lane = threadIdx.x & 31;
    if (node >= n) return;

    int beg = rowptr[node], end = rowptr[node + 1];
    int c = lane * 4;
    bool act = (c < d);

    float s = dinv[node];
    float s2 = s * s;
    float4 acc = make_float4(0.f, 0.f, 0.f, 0.f);
    if (act) {
        float4 hv = *(const float4*)(h + (size_t)node * d + c);
        acc.x = s2 * hv.x; acc.y = s2 * hv.y; acc.z = s2 * hv.z; acc.w = s2 * hv.w;
    }

    for (int jb = beg; jb < end; jb += 32) {
        int j = jb + lane;
        int t = 0; float cf = 0.0f;
        if (j < end) { t = eDst[j]; cf = eCoef[j]; }  // coalesced
        int m = end - jb; if (m > 32) m = 32;
        for (int k = 0; k < m; ++k) {
            int   tt = __shfl(t, k, 32);
            float cc = __shfl(cf, k, 32);
            if (act) {
                float4 hv = *(const float4*)(h + (size_t)tt * d + c);
                acc.x += cc * hv.x; acc.y += cc * hv.y;
                acc.z += cc * hv.z; acc.w += cc * hv.w;
            }
        }
    }
    if (act) *(float4*)(out + (size_t)node * d + c) = acc;
}

// ---------------------------------------------------------------------------
extern "C" void kernel_launch(void* const* d_in, const int* in_sizes, int n_in,
                              void* d_out, int out_size, void* d_ws, size_t ws_size,
                              hipStream_t stream) {
    const float* x  = (const float*)d_in[0];
    const int*   ei = (const int*)d_in[1];          // [2, NE] int32 per harness contract
    const float* ew = (const float*)d_in[2];
    const float* W1 = (const float*)d_in[3];
    const float* b1 = (const float*)d_in[4];
    const float* W2 = (const float*)d_in[5];
    const float* b2 = (const float*)d_in[6];
    const float* W3 = (const float*)d_in[7];
    const float* b3 = (const float*)d_in[8];
    float* out = (float*)d_out;

    const int* src = ei;
    const int* dst = ei + NE;

    // Workspace layout (256B-aligned slots)
    char* ws = (char*)d_ws;
    size_t off = 0;
    float* dinv   = (float*)(ws + off); off += 256 * 1024;           // NN floats
    int*   rowptr = (int*)  (ws + off); off += 256 * 1024;           // NN+1 ints
    int*   fill   = (int*)  (ws + off); off += 256 * 1024;           // NN ints (hist + cursor)
    int*   bsums  = (int*)  (ws + off); off += 4 * 1024;             // <=256 ints
    int*   eDst   = (int*)  (ws + off); off += (size_t)NE * 4;       // CSR neighbor ids
    float* eCoef  = (float*)(ws + off); off += (size_t)NE * 4;       // CSR edge coefficients
    float* bufA   = (float*)(ws + off); off += (size_t)NN * DHID * 4;
    float* bufB   = (float*)(ws + off);

    const int nb256 = (NN + 255) / 256;  // 196 scan blocks

    // --- degrees & d^-1/2 ---
    deg_init_kernel<<<nb256, 256, 0, stream>>>(dinv, NN);
    deg_accum_kernel<<<(NE + 255) / 256, 256, 0, stream>>>(src, ew, dinv, NE);
    dinv_kernel<<<nb256, 256, 0, stream>>>(dinv, NN);

    // --- CSR build (once, amortized over 3 layers) ---
    zero_int_kernel<<<nb256, 256, 0, stream>>>(fill, NN);
    hist_kernel<<<(NE + 255) / 256, 256, 0, stream>>>(src, fill, NE);
    scan_blocks_kernel<<<nb256, 256, 0, stream>>>(fill, rowptr, bsums, NN);
    scan_sums_kernel<<<1, 256, 0, stream>>>(bsums, nb256);
    add_off_kernel<<<nb256, 256, 0, stream>>>(rowptr, fill, bsums, NN, NE);
    scatter_kernel<<<(NE + 255) / 256, 256, 0, stream>>>(src, dst, ew, dinv, fill, eDst, eCoef, NE);

    const int tilesM = NN / 16;            // 3125, exact
    const int aggBlocks = (NN * 32 + 255) / 256;

    // --- layer 1: h1 = x @ W1^T + b1 ; out1 = dinv^2*h1 + gather (relu deferred) ---
    {
        int tn = DHID / 16, tot = tilesM * tn;
        gemm_wmma_kernel<false><<<(tot + 3) / 4, 128, 0, stream>>>(x, W1, b1, bufA, DIN, DHID, tot, tn);
        agg_gather_kernel<<<aggBlocks, 256, 0, stream>>>(rowptr, eDst, eCoef, dinv, bufA, bufB, NN, DHID);
    }
    // --- layer 2 ---
    {
        int tn = DHID / 16, tot = tilesM * tn;
        gemm_wmma_kernel<true><<<(tot + 3) / 4, 128, 0, stream>>>(bufB, W2, b2, bufA, DHID, DHID, tot, tn);
        agg_gather_kernel<<<aggBlocks, 256, 0, stream>>>(rowptr, eDst, eCoef, dinv, bufA, bufB, NN, DHID);
    }
    // --- layer 3 (no activation) -> d_out ---
    {
        int tn = DOUT / 16, tot = tilesM * tn;
        gemm_wmma_kernel<true><<<(tot + 3) / 4, 128, 0, stream>>>(bufB, W3, b3, bufA, DHID, DOUT, tot, tn);
        agg_gather_kernel<<<aggBlocks, 256, 0, stream>>>(rowptr, eDst, eCoef, dinv, bufA, out, NN, DOUT);
    }
}